// SlotAttention_6923487281185
// MI455X (gfx1250) — compile-verified
//
#include <hip/hip_runtime.h>

// ---------------------------------------------------------------------------
// Slot Attention on MI455X (gfx1250, wave32).
//   N = 262144 tokens, C = 64, K = 3 slots, HID = 128, 2 iterations.
// - k/v projection: V_WMMA_F32_16X16X4_F32 (fp32-exact; workload is
//   memory-bound at ~21 FLOP/B for the GEMM, <1 FLOP/B for the sweeps).
// - Tile staging: GLOBAL_LOAD_ASYNC_TO_LDS_B128 (ASYNCcnt) so the v-tile DMA
//   overlaps the logits/softmax phase.
// ---------------------------------------------------------------------------

typedef float v2f __attribute__((ext_vector_type(2)));
typedef float v8f __attribute__((ext_vector_type(8)));

#define ROWS_KV 128
#define ROWS_AT 128
#define XPITCH  68   // LDS row pitch in floats: 16B-aligned, low bank conflicts

// ---- gfx1250 async global->LDS copy (per-lane 16B), tracked by ASYNCcnt ----
__device__ __forceinline__ void async_load_b128(unsigned lds_off,
                                                const void* gbase,
                                                unsigned byte_off) {
    asm volatile("global_load_async_to_lds_b128 %0, %1, %2"
                 :: "v"(lds_off), "v"(byte_off), "s"(gbase)
                 : "memory");
}
__device__ __forceinline__ void wait_async0() {
    asm volatile("s_wait_asynccnt 0x0" ::: "memory");
}
__device__ __forceinline__ unsigned lds_addr_of(const void* p) {
    return (unsigned)(unsigned long long)p;   // low 32 bits = LDS offset
}

// ===========================================================================
// Kernel 1: fused LayerNorm(x) -> k = xn@Wk.T, v = xn@Wv.T  (WMMA f32 16x16x4)
// ===========================================================================
__global__ __launch_bounds__(256) void kv_kernel(
    const float* __restrict__ x,
    const float* __restrict__ ln_g, const float* __restrict__ ln_b,
    const float* __restrict__ Wk,   const float* __restrict__ Wv,
    float* __restrict__ kout, float* __restrict__ vout, int N)
{
    __shared__ float xn[ROWS_KV * XPITCH];
    const int  tid  = threadIdx.x;
    const long row0 = (long)blockIdx.x * ROWS_KV;

    // ---- async-DMA the 128x64 x-tile into LDS (16B per lane per issue) ----
    const unsigned xn_base = lds_addr_of(xn);
    #pragma unroll
    for (int i = 0; i < 8; ++i) {
        int li = tid + 256 * i;            // 0..2047 float4 slots
        int r  = li >> 4, c4 = li & 15;
        async_load_b128(xn_base + (unsigned)(r * XPITCH + c4 * 4) * 4u,
                        x, (unsigned)((row0 + r) * 64 + c4 * 4) * 4u);
    }
    wait_async0();
    __syncthreads();

    // ---- LayerNorm, one thread per row (threads 0..127) ----
    if (tid < ROWS_KV) {
        float* rp = &xn[tid * XPITCH];
        float m = 0.f;
        #pragma unroll
        for (int c = 0; c < 64; ++c) m += rp[c];
        m *= (1.0f / 64.0f);
        float var = 0.f;
        #pragma unroll
        for (int c = 0; c < 64; ++c) { float d = rp[c] - m; var += d * d; }
        float rstd = rsqrtf(var * (1.0f / 64.0f) + 1e-5f);
        #pragma unroll
        for (int c = 0; c < 64; ++c) rp[c] = (rp[c] - m) * rstd * ln_g[c] + ln_b[c];
    }
    __syncthreads();

    // ---- WMMA GEMM: each wave owns one 16-row tile of the output ----
    const int wave = tid >> 5, lane = tid & 31;
    const int mrow = wave * 16;          // row tile base within block
    const int ml   = lane & 15;          // M (for A) / N (for B) within tile
    const int hi   = lane >> 4;          // lane half selects K pair {0,1}/{2,3}
    for (int mat = 0; mat < 2; ++mat) {
        const float* __restrict__ W   = mat ? Wv   : Wk;
        float*       __restrict__ out = mat ? vout : kout;
        for (int nt = 0; nt < 4; ++nt) {
            v8f acc = {};
            #pragma unroll
            for (int kc = 0; kc < 16; ++kc) {
                // A fragment 16x4: lane<16 -> K={0,1}, lane>=16 -> K={2,3}
                const float* ap = &xn[(mrow + ml) * XPITCH + kc * 4 + hi * 2];
                v2f a; a.x = ap[0]; a.y = ap[1];
                // B fragment 4x16: B[kk,n] = W.T[4kc+kk, 16nt+n] = W[16nt+n, 4kc+kk]
                const float* bp = &W[(nt * 16 + ml) * 64 + kc * 4 + hi * 2];
                v2f b; b.x = bp[0]; b.y = bp[1];
                acc = __builtin_amdgcn_wmma_f32_16x16x4_f32(
                          false, a, false, b, (short)0, acc, false, false);
            }
            // D 16x16: VGPR j holds rows {j, j+8}; lane%16 is the column
            float* op = out + (row0 + mrow + hi * 8) * 64 + nt * 16 + ml;
            #pragma unroll
            for (int j = 0; j < 8; ++j) op[(long)j * 64] = acc[j];
        }
    }
}

// ===========================================================================
// Kernel 2: attention sweep. logits -> softmax(3) -> unnormalized w.T @ v
// accumulated into global [3,64] + per-slot sums (normalization is a per-slot
// scalar divide done later: (w.T@v)/sum(w) == normalized-w.T @ v).
// The v-tile is DMA'd to LDS asynchronously and overlaps the logits phase.
// ===========================================================================
__global__ __launch_bounds__(256) void attn_kernel(
    const float* __restrict__ kbuf, const float* __restrict__ vbuf,
    const float* __restrict__ q,    // [3,64]
    float* __restrict__ upd, float* __restrict__ sums,
    float* __restrict__ attn_out,   // [3,N] or nullptr
    int N)
{
    __shared__ float lv[ROWS_AT * 64];
    __shared__ float lw[3 * ROWS_AT];
    __shared__ float lq[192];
    const int  tid  = threadIdx.x;
    const long row0 = (long)blockIdx.x * ROWS_AT;

    // ---- kick off async v-tile DMA (32 KB); completes during logits ----
    const unsigned lv_base = lds_addr_of(lv);
    #pragma unroll
    for (int i = 0; i < 8; ++i) {
        int li = tid + 256 * i;
        async_load_b128(lv_base + (unsigned)li * 16u,
                        vbuf, (unsigned)(row0 * 64) * 4u + (unsigned)li * 16u);
    }
    if (tid < 192) lq[tid] = q[tid];
    __syncthreads();                       // lq visible; v DMA still in flight

    if (tid < ROWS_AT) {
        const long row = row0 + tid;
        const float4* k4 = reinterpret_cast<const float4*>(kbuf + row * 64);
        float l0 = 0.f, l1 = 0.f, l2 = 0.f;
        #pragma unroll
        for (int c4 = 0; c4 < 16; ++c4) {
            float4 kk = k4[c4]; int c = c4 * 4;
            l0 += kk.x*lq[c]      + kk.y*lq[c+1]      + kk.z*lq[c+2]      + kk.w*lq[c+3];
            l1 += kk.x*lq[64+c]   + kk.y*lq[64+c+1]   + kk.z*lq[64+c+2]   + kk.w*lq[64+c+3];
            l2 += kk.x*lq[128+c]  + kk.y*lq[128+c+1]  + kk.z*lq[128+c+2]  + kk.w*lq[128+c+3];
        }
        float mx = fmaxf(l0, fmaxf(l1, l2));
        float e0 = __expf(l0 - mx), e1 = __expf(l1 - mx), e2 = __expf(l2 - mx);
        float inv = 1.0f / (e0 + e1 + e2);
        float a0 = e0 * inv, a1 = e1 * inv, a2 = e2 * inv;
        if (attn_out) {
            attn_out[row]            = a0;
            attn_out[(long)N + row]  = a1;
            attn_out[2L * N + row]   = a2;
        }
        lw[tid]             = a0 + 1e-8f;
        lw[ROWS_AT + tid]   = a1 + 1e-8f;
        lw[2*ROWS_AT + tid] = a2 + 1e-8f;
    }
    wait_async0();                         // v tile resident
    __syncthreads();

    if (tid < 192) {            // thread (s,c) reduces one output element
        int s = tid >> 6, c = tid & 63;
        const float* wr = &lw[s * ROWS_AT];
        float acc = 0.f;
        for (int r = 0; r < ROWS_AT; ++r) acc += wr[r] * lv[r * 64 + c];
        atomicAdd(&upd[s * 64 + c], acc);
    }
    if (tid < 3) {
        float acc = 0.f;
        for (int r = 0; r < ROWS_AT; ++r) acc += lw[tid * ROWS_AT + r];
        atomicAdd(&sums[tid], acc);
    }
}

// ===========================================================================
// Kernel 3: slot-side update (single block). phase 0: init slots + q + zero
// accumulators. phase 1: GRU+MLP, new q, re-zero accumulators. phase 2:
// GRU+MLP, write final slots.
// ===========================================================================
struct SlotParams {
    const float *noise_fg, *noise_bg, *mu_fg, *logsig_fg, *mu_bg, *logsig_bg;
    const float *lnq_fg_g, *lnq_fg_b, *Wq_fg, *lnq_bg_g, *lnq_bg_b, *Wq_bg;
    const float *gfg_wih, *gfg_whh, *gfg_bih, *gfg_bhh;
    const float *gbg_wih, *gbg_whh, *gbg_bih, *gbg_bhh;
    const float *mfg_ln_g, *mfg_ln_b, *mfg_w1, *mfg_b1, *mfg_w2, *mfg_b2;
    const float *mbg_ln_g, *mbg_ln_b, *mbg_w1, *mbg_b1, *mbg_w2, *mbg_b2;
    float *slot, *q, *upd, *sums, *out_slots;
    int phase;
};

__global__ __launch_bounds__(256) void slots_kernel(SlotParams P)
{
    __shared__ float lh[192], lx[192], lgi[576], lgh[576];
    __shared__ float lhn[192], lxn[192], lhid[384], lstat[8];
    const int tid = threadIdx.x;
    const int s   = (tid & 255) >> 6;   // slot index for tid<192 (0=bg,1,2=fg)
    const int c   = tid & 63;

    if (P.phase == 0) {
        if (tid < 192) {
            float val = (s == 0)
                ? P.mu_bg[c] + __expf(P.logsig_bg[c]) * P.noise_bg[c]
                : P.mu_fg[c] + __expf(P.logsig_fg[c]) * P.noise_fg[(s-1)*64 + c];
            P.slot[tid] = val;
            lh[tid] = val;
        }
    } else {
        if (tid < 192) {
            lh[tid] = P.slot[tid];
            lx[tid] = P.upd[tid] / P.sums[s];
        }
        __syncthreads();
        // GRU gates: gi = x@wih.T+bih, gh = h@whh.T+bhh  (3 slots x 192 outs)
        for (int idx = tid; idx < 576; idx += 256) {
            int ss = idx / 192, j = idx % 192;
            const float* wih = ss ? P.gfg_wih : P.gbg_wih;
            const float* whh = ss ? P.gfg_whh : P.gbg_whh;
            float gi = (ss ? P.gfg_bih : P.gbg_bih)[j];
            float gh = (ss ? P.gfg_bhh : P.gbg_bhh)[j];
            for (int cc = 0; cc < 64; ++cc) {
                gi += wih[j*64+cc] * lx[ss*64+cc];
                gh += whh[j*64+cc] * lh[ss*64+cc];
            }
            lgi[idx] = gi; lgh[idx] = gh;
        }
        __syncthreads();
        if (tid < 192) {
            float ir = lgi[s*192+c],     hr = lgh[s*192+c];
            float iz = lgi[s*192+64+c],  hz = lgh[s*192+64+c];
            float in_= lgi[s*192+128+c], hn = lgh[s*192+128+c];
            float r = 1.0f / (1.0f + __expf(-(ir + hr)));
            float z = 1.0f / (1.0f + __expf(-(iz + hz)));
            float n = tanhf(in_ + r * hn);
            lhn[tid] = (1.0f - z) * n + z * lh[tid];
        }
        __syncthreads();
        // residual MLP: LN -> relu(w1) -> w2
        if (tid < 3) {
            float m = 0.f;
            for (int cc = 0; cc < 64; ++cc) m += lhn[tid*64+cc];
            m *= (1.0f/64.f);
            float v = 0.f;
            for (int cc = 0; cc < 64; ++cc) { float d = lhn[tid*64+cc]-m; v += d*d; }
            lstat[tid] = m; lstat[4+tid] = rsqrtf(v*(1.0f/64.f) + 1e-5f);
        }
        __syncthreads();
        if (tid < 192) {
            const float* g = s ? P.mfg_ln_g : P.mbg_ln_g;
            const float* b = s ? P.mfg_ln_b : P.mbg_ln_b;
            lxn[tid] = (lhn[tid] - lstat[s]) * lstat[4+s] * g[c] + b[c];
        }
        __syncthreads();
        for (int idx = tid; idx < 384; idx += 256) {
            int ss = idx / 128, j = idx % 128;
            const float* w1 = ss ? P.mfg_w1 : P.mbg_w1;
            float acc = (ss ? P.mfg_b1 : P.mbg_b1)[j];
            for (int cc = 0; cc < 64; ++cc) acc += w1[j*64+cc] * lxn[ss*64+cc];
            lhid[idx] = fmaxf(acc, 0.0f);
        }
        __syncthreads();
        if (tid < 192) {
            const float* w2 = s ? P.mfg_w2 : P.mbg_w2;
            float acc = (s ? P.mfg_b2 : P.mbg_b2)[c];
            for (int jj = 0; jj < 128; ++jj) acc += w2[c*128+jj] * lhid[s*128+jj];
            float ns = lhn[tid] + acc;
            P.slot[tid] = ns;
            lh[tid] = ns;
            if (P.phase == 2) P.out_slots[tid] = ns;   // [bg, fg0, fg1] order
        }
    }
    __syncthreads();

    if (P.phase != 2) {
        // q = (LN(slot) @ Wq.T) * SLOT_DIM^-0.5 ; reset accumulators
        if (tid < 3) {
            float m = 0.f;
            for (int cc = 0; cc < 64; ++cc) m += lh[tid*64+cc];
            m *= (1.0f/64.f);
            float v = 0.f;
            for (int cc = 0; cc < 64; ++cc) { float d = lh[tid*64+cc]-m; v += d*d; }
            lstat[tid] = m; lstat[4+tid] = rsqrtf(v*(1.0f/64.f) + 1e-5f);
        }
        __syncthreads();
        if (tid < 192) {
            const float* g = s ? P.lnq_fg_g : P.lnq_bg_g;
            const float* b = s ? P.lnq_fg_b : P.lnq_bg_b;
            lxn[tid] = (lh[tid] - lstat[s]) * lstat[4+s] * g[c] + b[c];
        }
        __syncthreads();
        if (tid < 192) {
            const float* Wq = s ? P.Wq_fg : P.Wq_bg;
            float acc = 0.f;
            for (int cc = 0; cc < 64; ++cc) acc += Wq[c*64+cc] * lxn[s*64+cc];
            P.q[tid]   = acc * 0.125f;   // 64^-0.5
            P.upd[tid] = 0.0f;
        }
        if (tid < 3) P.sums[tid] = 0.0f;
    }
}

// ===========================================================================
extern "C" void kernel_launch(void* const* d_in, const int* in_sizes, int n_in,
                              void* d_out, int out_size, void* d_ws, size_t ws_size,
                              hipStream_t stream)
{
    const float* x       = (const float*)d_in[0];
    const float* ln_in_g = (const float*)d_in[3];
    const float* ln_in_b = (const float*)d_in[4];
    const float* Wk      = (const float*)d_in[9];
    const float* Wv      = (const float*)d_in[10];
    const int N = in_sizes[0] / 64;

    float* ws    = (float*)d_ws;
    float* kbuf  = ws;
    float* vbuf  = kbuf + (size_t)N * 64;
    float* slot  = vbuf + (size_t)N * 64;
    float* qbuf  = slot + 192;
    float* upd   = qbuf + 192;
    float* sums  = upd + 192;

    float* out_slots = (float*)d_out;        // [3,64]
    float* out_attn  = out_slots + 192;      // [3,N]

    SlotParams P;
    P.noise_fg = (const float*)d_in[1];  P.noise_bg  = (const float*)d_in[2];
    P.mu_fg    = (const float*)d_in[5];  P.logsig_fg = (const float*)d_in[6];
    P.mu_bg    = (const float*)d_in[7];  P.logsig_bg = (const float*)d_in[8];
    P.lnq_fg_g = (const float*)d_in[11]; P.lnq_fg_b  = (const float*)d_in[12];
    P.Wq_fg    = (const float*)d_in[13];
    P.lnq_bg_g = (const float*)d_in[14]; P.lnq_bg_b  = (const float*)d_in[15];
    P.Wq_bg    = (const float*)d_in[16];
    P.gfg_wih  = (const float*)d_in[17]; P.gfg_whh = (const float*)d_in[18];
    P.gfg_bih  = (const float*)d_in[19]; P.gfg_bhh = (const float*)d_in[20];
    P.gbg_wih  = (const float*)d_in[21]; P.gbg_whh = (const float*)d_in[22];
    P.gbg_bih  = (const float*)d_in[23]; P.gbg_bhh = (const float*)d_in[24];
    P.mfg_ln_g = (const float*)d_in[25]; P.mfg_ln_b = (const float*)d_in[26];
    P.mfg_w1   = (const float*)d_in[27]; P.mfg_b1   = (const float*)d_in[28];
    P.mfg_w2   = (const float*)d_in[29]; P.mfg_b2   = (const float*)d_in[30];
    P.mbg_ln_g = (const float*)d_in[31]; P.mbg_ln_b = (const float*)d_in[32];
    P.mbg_w1   = (const float*)d_in[33]; P.mbg_b1   = (const float*)d_in[34];
    P.mbg_w2   = (const float*)d_in[35]; P.mbg_b2   = (const float*)d_in[36];
    P.slot = slot; P.q = qbuf; P.upd = upd; P.sums = sums; P.out_slots = out_slots;

    const int kvBlocks = N / ROWS_KV;
    const int atBlocks = N / ROWS_AT;

    P.phase = 0;
    slots_kernel<<<1, 256, 0, stream>>>(P);                       // init + q0
    kv_kernel<<<kvBlocks, 256, 0, stream>>>(x, ln_in_g, ln_in_b,  // LN + WMMA
                                            Wk, Wv, kbuf, vbuf, N);
    attn_kernel<<<atBlocks, 256, 0, stream>>>(kbuf, vbuf, qbuf,   // iter 0
                                              upd, sums, nullptr, N);
    P.phase = 1;
    slots_kernel<<<1, 256, 0, stream>>>(P);                       // update + q1
    attn_kernel<<<atBlocks, 256, 0, stream>>>(kbuf, vbuf, qbuf,   // iter 1
                                              upd, sums, out_attn, N);
    P.phase = 2;
    slots_kernel<<<1, 256, 0, stream>>>(P);                       // final slots
}